// LQRSim_71717363908918
// MI455X (gfx1250) — compile-verified
//
#include <hip/hip_runtime.h>
#include <stdint.h>

#define SN 256
#define SM 128
#define NM 384
#define ST 1024

typedef __attribute__((ext_vector_type(2))) float v2f;
typedef __attribute__((ext_vector_type(4))) float f4;
typedef __attribute__((ext_vector_type(8))) float v8f;

// Async global->LDS copy of 16B; INST_OFFSET advances both global and LDS addr.
#define ASYNC_CP128(ldsaddr, voff, sbase, OFF)                                  \
    asm volatile("global_load_async_to_lds_b128 %0, %1, %2 offset:" #OFF        \
                 :                                                              \
                 : "v"(ldsaddr), "v"(voff), "s"(sbase)                          \
                 : "memory")

#define WAIT_ASYNC0() asm volatile("s_wait_asynccnt 0x0" ::: "memory")

// ---------------------------------------------------------------------------
// Phase A: sequential rollout. One workgroup (1024 threads = 32 wave32) on one
// WGP. F lives in registers (96 floats/thread). K_t / k_t are double-buffered
// in LDS via ASYNCcnt-tracked global_load_async_to_lds_b128, staged one full
// step ahead so no global-load latency sits on the dependency chain.
// ---------------------------------------------------------------------------
__global__ __launch_bounds__(1024, 1) void lqr_rollout(
    const float* __restrict__ F, const float* __restrict__ f,
    const float* __restrict__ Ks, const float* __restrict__ ks,
    const float* __restrict__ x0,
    float* __restrict__ states, float* __restrict__ actions)
{
    __shared__ float Kbuf[2][SM * SN];   // 2 x 128 KB
    __shared__ float ksbuf[2][SM];       // 2 x 512 B
    __shared__ float inp[2][NM];         // [x(256) ; u(128)] double buffered

    const int tid = threadIdx.x;

    // x' mapping: 4 threads per output row, 96 elements each
    const int o   = tid >> 2;            // 0..255
    const int sub = tid & 3;             // 0..3
    // u mapping: 8 threads per action row, 32 elements each
    const int a   = tid >> 3;            // 0..127
    const int seg = tid & 7;             // 0..7

    // Persistent register copy of this thread's F slice (row o, cols sub*96..)
    float Freg[96];
    {
        const float* Frow = F + (size_t)o * NM + sub * 96;
        #pragma unroll
        for (int i = 0; i < 96; i += 4) {
            f4 v = *(const f4*)(Frow + i);
            Freg[i]   = v.x; Freg[i+1] = v.y;
            Freg[i+2] = v.z; Freg[i+3] = v.w;
        }
    }
    const float fo = f[o];

    // Stage K_t (+ k_t) into LDS buffer b: each thread copies 128 B (8 x b128).
    auto stageK = [&](int t, int b) {
        const float* sbase = Ks + (size_t)t * SM * SN;
        const unsigned lds = (unsigned)(uintptr_t)&Kbuf[b][tid * 32];
        const int voff = tid * 128;
        ASYNC_CP128(lds, voff, sbase, 0);
        ASYNC_CP128(lds, voff, sbase, 16);
        ASYNC_CP128(lds, voff, sbase, 32);
        ASYNC_CP128(lds, voff, sbase, 48);
        ASYNC_CP128(lds, voff, sbase, 64);
        ASYNC_CP128(lds, voff, sbase, 80);
        ASYNC_CP128(lds, voff, sbase, 96);
        ASYNC_CP128(lds, voff, sbase, 112);
        if (tid < 32) {                      // 32 lanes x 16 B = 512 B = k_t
            const float* kb = ks + (size_t)t * SM;
            const unsigned lk = (unsigned)(uintptr_t)&ksbuf[b][tid * 4];
            const int vo2 = tid * 16;
            ASYNC_CP128(lk, vo2, kb, 0);
        }
    };

    stageK(0, 0);
    if (tid < SN) { inp[0][tid] = x0[tid]; states[tid] = x0[tid]; }
    WAIT_ASYNC0();
    __syncthreads();

    for (int t = 0; t < ST; ++t) {
        const int cur = t & 1, nxt = cur ^ 1;

        // Kick off next step's K/k copy; it has the whole step to land.
        if (t + 1 < ST) stageK(t + 1, nxt);

        // ---------------- u = K_t x + k_t (all from LDS) ----------------
        const float* Kp = &Kbuf[cur][a * SN + seg * 32];
        const float* xp = &inp[cur][seg * 32];
        float acc0 = 0.f, acc1 = 0.f;
        #pragma unroll
        for (int i = 0; i < 32; i += 8) {
            f4 kv0 = *(const f4*)(Kp + i);
            f4 xv0 = *(const f4*)(xp + i);
            f4 kv1 = *(const f4*)(Kp + i + 4);
            f4 xv1 = *(const f4*)(xp + i + 4);
            acc0 = fmaf(kv0.x, xv0.x, fmaf(kv0.y, xv0.y,
                   fmaf(kv0.z, xv0.z, fmaf(kv0.w, xv0.w, acc0))));
            acc1 = fmaf(kv1.x, xv1.x, fmaf(kv1.y, xv1.y,
                   fmaf(kv1.z, xv1.z, fmaf(kv1.w, xv1.w, acc1))));
        }
        float s = acc0 + acc1;
        s += __shfl_xor(s, 1);
        s += __shfl_xor(s, 2);
        s += __shfl_xor(s, 4);
        if (seg == 0) {
            float u = s + ksbuf[cur][a];
            inp[cur][SN + a] = u;
            actions[(size_t)t * SM + a] = u;
        }
        __syncthreads();

        // ---------------- x' = F [x;u] + f ----------------
        const float* ip = &inp[cur][sub * 96];
        float b0 = 0.f, b1 = 0.f, b2 = 0.f, b3 = 0.f;
        #pragma unroll
        for (int i = 0; i < 96; i += 4) {
            f4 iv = *(const f4*)(ip + i);
            b0 = fmaf(Freg[i],     iv.x, b0);
            b1 = fmaf(Freg[i + 1], iv.y, b1);
            b2 = fmaf(Freg[i + 2], iv.z, b2);
            b3 = fmaf(Freg[i + 3], iv.w, b3);
        }
        float xs = (b0 + b1) + (b2 + b3);
        xs += __shfl_xor(xs, 1);
        xs += __shfl_xor(xs, 2);
        if (sub == 0) {
            float xn = xs + fo;
            inp[nxt][o] = xn;
            states[(size_t)(t + 1) * SN + o] = xn;
        }
        // Ensure next step's async K copy has landed before crossing barrier.
        WAIT_ASYNC0();
        __syncthreads();
    }
}

// ---------------------------------------------------------------------------
// Phase B: batched quadratic costs via f32 WMMA.
// One wave per block handles 16 timesteps: Z = Inp[16x384] * C[384x384]
// accumulated 16x16x4 at a time, cost[t] = sum_n Inp[t,n]*(0.5*Z[t,n] + c[n]).
// ---------------------------------------------------------------------------
__global__ __launch_bounds__(32) void lqr_costs(
    const float* __restrict__ C, const float* __restrict__ c,
    const float* __restrict__ states, const float* __restrict__ actions,
    float* __restrict__ costs)
{
    __shared__ float sInp[16 * 388];       // 16 rows, padded stride (bank spread)
    const int lane = threadIdx.x;          // 0..31
    const int t0   = blockIdx.x * 16;

    // Cooperative load of the 16x384 Inp tile (rows t0..t0+15), zero padded.
    for (int idx = lane; idx < 16 * 96; idx += 32) {   // 96 float4 per row
        const int r = idx / 96, q = idx % 96;
        const int row = t0 + r, col = q * 4;
        f4 v = {0.f, 0.f, 0.f, 0.f};
        if (row <= ST) {
            if (col < SN)       v = *(const f4*)(states + (size_t)row * SN + col);
            else if (row < ST)  v = *(const f4*)(actions + (size_t)row * SM + (col - SN));
        }
        float* dst = &sInp[r * 388 + col];
        dst[0] = v.x; dst[1] = v.y; dst[2] = v.z; dst[3] = v.w;
    }
    __syncthreads();

    const int half = lane >> 4;            // 0: lanes 0-15, 1: lanes 16-31
    const int l15  = lane & 15;

    float rowacc[8];
    #pragma unroll
    for (int j = 0; j < 8; ++j) rowacc[j] = 0.f;

    for (int n0 = 0; n0 < NM; n0 += 16) {
        v8f d = {0.f, 0.f, 0.f, 0.f, 0.f, 0.f, 0.f, 0.f};
        for (int k0 = 0; k0 < NM; k0 += 4) {
            const int ka = k0 + 2 * half;
            v2f aa, bb;
            // A 16x4 layout: VGPR0 K={0|2}, VGPR1 K={1|3}; row = lane&15
            aa.x = sInp[l15 * 388 + ka];
            aa.y = sInp[l15 * 388 + ka + 1];
            // B 4x16 layout: VGPR0 K={0|2}, VGPR1 K={1|3}; col = lane&15
            bb.x = C[(size_t)ka * NM + n0 + l15];
            bb.y = C[(size_t)(ka + 1) * NM + n0 + l15];
            d = __builtin_amdgcn_wmma_f32_16x16x4_f32(
                    false, aa, false, bb, (short)0, d, false, false);
        }
        const float cl = c[n0 + l15];
        #pragma unroll
        for (int j = 0; j < 8; ++j) {
            // D layout: VGPR j -> row t0+j (lanes 0-15) / t0+j+8 (lanes 16-31)
            const float iv = sInp[(j + 8 * half) * 388 + n0 + l15];
            rowacc[j] = fmaf(iv, 0.5f * d[j] + cl, rowacc[j]);
        }
    }

    #pragma unroll
    for (int j = 0; j < 8; ++j) {
        float s = rowacc[j];
        s += __shfl_xor(s, 1);
        s += __shfl_xor(s, 2);
        s += __shfl_xor(s, 4);
        s += __shfl_xor(s, 8);
        const int row = t0 + j + 8 * half;
        if (l15 == 0 && row <= ST) costs[row] = s;
    }
}

// ---------------------------------------------------------------------------
extern "C" void kernel_launch(void* const* d_in, const int* in_sizes, int n_in,
                              void* d_out, int out_size, void* d_ws, size_t ws_size,
                              hipStream_t stream) {
    const float* F  = (const float*)d_in[0];   // [256, 384]
    const float* f  = (const float*)d_in[1];   // [256]
    const float* C  = (const float*)d_in[2];   // [384, 384]
    const float* c  = (const float*)d_in[3];   // [384]
    const float* Ks = (const float*)d_in[4];   // [1024, 128, 256]
    const float* ks = (const float*)d_in[5];   // [1024, 128]
    const float* x0 = (const float*)d_in[6];   // [256]

    float* states  = (float*)d_out;                         // [1025, 256]
    float* actions = states + (size_t)(ST + 1) * SN;        // [1024, 128]
    float* costs   = actions + (size_t)ST * SM;             // [1025]

    lqr_rollout<<<1, 1024, 0, stream>>>(F, f, Ks, ks, x0, states, actions);
    lqr_costs<<<(ST + 16) / 16, 32, 0, stream>>>(C, c, states, actions, costs);
}